// Discriminator_3934190044271
// MI455X (gfx1250) — compile-verified
//
#include <hip/hip_runtime.h>

#define IN_DIM 256
#define HID 64

typedef __attribute__((ext_vector_type(2))) float v2f;
typedef __attribute__((ext_vector_type(8))) float v8f;

static __device__ __forceinline__ void wmma_f32_k4(v8f& acc, v2f a, v2f b) {
  // D = A(16x4 f32) * B(4x16 f32) + C ; args: (neg_a, A, neg_b, B, c_mod, C, reuse_a, reuse_b)
  acc = __builtin_amdgcn_wmma_f32_16x16x4_f32(false, a, false, b, (short)0, acc,
                                              false, false);
}

// ---- init: zero accumulators, deg=1.0 (self-loop) --------------------------
__global__ void k_init(float* hagg, float* deg, float* agg2, int N) {
  int t = blockIdx.x * blockDim.x + threadIdx.x;
  if (t < N * HID) hagg[t] = 0.0f;
  if (t < N) { deg[t] = 1.0f; agg2[t] = 0.0f; }
}

// ---- degree count over destination nodes -----------------------------------
__global__ void k_count_deg(const long long* __restrict__ dst, float* deg, int E) {
  int e = blockIdx.x * blockDim.x + threadIdx.x;
  if (e < E) atomicAdd(&deg[(size_t)dst[e]], 1.0f);
}

__global__ void k_dinv(const float* __restrict__ deg, float* dinv, int N) {
  int i = blockIdx.x * blockDim.x + threadIdx.x;
  if (i < N) dinv[i] = rsqrtf(deg[i]);
}

// ---- H[M,64] = X[M,256] @ W[256,64] via V_WMMA_F32_16X16X4_F32 -------------
// One wave computes a 16x64 output tile (4 accumulators), K swept in steps of 4.
// W1 (64KB) staged in LDS once per block, shared by 8 waves.
__global__ __launch_bounds__(256) void k_gemm(const float* __restrict__ X,
                                              const float* __restrict__ W,
                                              float* __restrict__ H, int M) {
  __shared__ float Wl[IN_DIM * HID];  // 64 KB of 320 KB/WGP
  for (int i = threadIdx.x; i < IN_DIM * HID; i += 256) Wl[i] = W[i];
  __syncthreads();

  const int wave = threadIdx.x >> 5;
  const int lane = threadIdx.x & 31;
  const int tile = blockIdx.x * 8 + wave;
  const int row0 = tile * 16;
  if (row0 >= M) return;

  const int r    = lane & 15;       // row (A) / col (B,D) within tile
  const int hi   = lane >> 4;       // half-wave select
  const int koff = hi * 2;          // A/B: hi half holds K+2,K+3

  const float* xrow = X + (size_t)(row0 + r) * IN_DIM;
  v8f acc0 = {}; v8f acc1 = {}; v8f acc2 = {}; v8f acc3 = {};

  for (int k = 0; k < IN_DIM; k += 4) {
    v2f a; a.x = xrow[k + koff]; a.y = xrow[k + koff + 1];
    const float* w0 = &Wl[(size_t)(k + koff) * HID];
    const float* w1 = &Wl[(size_t)(k + koff + 1) * HID];
    v2f b0; b0.x = w0[r];      b0.y = w1[r];
    v2f b1; b1.x = w0[16 + r]; b1.y = w1[16 + r];
    v2f b2; b2.x = w0[32 + r]; b2.y = w1[32 + r];
    v2f b3; b3.x = w0[48 + r]; b3.y = w1[48 + r];
    wmma_f32_k4(acc0, a, b0);
    wmma_f32_k4(acc1, a, b1);
    wmma_f32_k4(acc2, a, b2);
    wmma_f32_k4(acc3, a, b3);
  }

  // D layout: VGPR v -> row v + hi*8, col = r (+16*ctile)
  float* hp = H + (size_t)row0 * HID;
  for (int v = 0; v < 8; ++v) {
    int rr = v + hi * 8;
    hp[(size_t)rr * HID + r]      = acc0[v];
    hp[(size_t)rr * HID + 16 + r] = acc1[v];
    hp[(size_t)rr * HID + 32 + r] = acc2[v];
    hp[(size_t)rr * HID + 48 + r] = acc3[v];
  }
}

// ---- layer-1 edge aggregation: 64 threads (2 waves) per edge ---------------
// Edge id is wave-uniform -> scalarized index/dinv loads; 32 contiguous dims
// per wave -> fully coalesced 128B gather + atomic scatter.
__global__ void k_edge_agg1(const long long* __restrict__ src,
                            const long long* __restrict__ dst,
                            const float* __restrict__ dinv,
                            const float* __restrict__ h,
                            float* hagg, int E) {
  int t = blockIdx.x * blockDim.x + threadIdx.x;
  int e = __builtin_amdgcn_readfirstlane(t >> 6);
  if (e >= E) return;
  int d = t & 63;
  long long s  = src[e];
  long long dd = dst[e];
  float nw = dinv[s] * dinv[dd];
  atomicAdd(&hagg[(size_t)dd * HID + d], nw * h[(size_t)s * HID + d]);
}

// ---- add self-loop message + bias, ReLU (in place on hagg) -----------------
__global__ void k_selfloop_relu(const float* __restrict__ h,
                                const float* __restrict__ dinv,
                                const float* __restrict__ b1,
                                float* hagg, int N) {
  int t = blockIdx.x * blockDim.x + threadIdx.x;
  if (t >= N * HID) return;
  int i = t >> 6;
  int d = t & 63;
  float di = dinv[i];
  float v = hagg[t] + di * di * h[t] + b1[d];
  hagg[t] = fmaxf(v, 0.0f);
}

// ---- g[i] = dot(hagg[i,:], W2) : one wave32 per node, shfl reduction -------
__global__ void k_dot_w2(const float* __restrict__ hagg,
                         const float* __restrict__ W2,
                         float* g, int N) {
  int wave = threadIdx.x >> 5;
  int lane = threadIdx.x & 31;
  int i = blockIdx.x * (blockDim.x >> 5) + wave;
  if (i >= N) return;
  const float* hp = hagg + (size_t)i * HID;
  float p = hp[lane] * W2[lane] + hp[lane + 32] * W2[lane + 32];
  for (int off = 16; off > 0; off >>= 1) p += __shfl_xor(p, off, 32);
  if (lane == 0) g[i] = p;
}

// ---- layer-2 edge aggregation (scalar per edge) ----------------------------
__global__ void k_edge_agg2(const long long* __restrict__ src,
                            const long long* __restrict__ dst,
                            const float* __restrict__ dinv,
                            const float* __restrict__ g,
                            float* agg2, int E) {
  int e = blockIdx.x * blockDim.x + threadIdx.x;
  if (e >= E) return;
  long long s  = src[e];
  long long dd = dst[e];
  atomicAdd(&agg2[dd], dinv[s] * dinv[dd] * g[s]);
}

// ---- self-loop + bias + sigmoid -------------------------------------------
__global__ void k_final(const float* __restrict__ agg2,
                        const float* __restrict__ dinv,
                        const float* __restrict__ g,
                        const float* __restrict__ b2,
                        float* out, int N) {
  int i = blockIdx.x * blockDim.x + threadIdx.x;
  if (i >= N) return;
  float v = agg2[i] + dinv[i] * dinv[i] * g[i] + b2[0];
  out[i] = 1.0f / (1.0f + expf(-v));
}

extern "C" void kernel_launch(void* const* d_in, const int* in_sizes, int n_in,
                              void* d_out, int out_size, void* d_ws, size_t ws_size,
                              hipStream_t stream) {
  const float*     x   = (const float*)d_in[0];
  const long long* ei  = (const long long*)d_in[1];  // int64 edge_index [2,E]
  const float*     W1  = (const float*)d_in[2];
  const float*     b1  = (const float*)d_in[3];
  const float*     W2  = (const float*)d_in[4];
  const float*     b2  = (const float*)d_in[5];
  float*           out = (float*)d_out;

  const int N = in_sizes[0] / IN_DIM;
  const int E = in_sizes[1] / 2;
  const long long* src = ei;
  const long long* dst = ei + E;

  // workspace layout (floats): h[N*64] | hagg[N*64] | deg[N] | dinv[N] | g[N] | agg2[N]
  float* ws   = (float*)d_ws;
  float* h    = ws;
  float* hagg = h    + (size_t)N * HID;
  float* deg  = hagg + (size_t)N * HID;
  float* dinv = deg  + N;
  float* g    = dinv + N;
  float* agg2 = g    + N;

  const int NH = N * HID;
  k_init<<<(NH + 255) / 256, 256, 0, stream>>>(hagg, deg, agg2, N);
  k_count_deg<<<(E + 255) / 256, 256, 0, stream>>>(dst, deg, E);
  k_dinv<<<(N + 255) / 256, 256, 0, stream>>>(deg, dinv, N);

  const int tiles = (N + 15) / 16;
  k_gemm<<<(tiles + 7) / 8, 256, 0, stream>>>(x, W1, h, N);

  long long ethreads = (long long)E * HID;
  k_edge_agg1<<<(int)((ethreads + 255) / 256), 256, 0, stream>>>(src, dst, dinv, h, hagg, E);
  k_selfloop_relu<<<(NH + 255) / 256, 256, 0, stream>>>(h, dinv, b1, hagg, N);

  k_dot_w2<<<(N + 7) / 8, 256, 0, stream>>>(hagg, W2, g, N);
  k_edge_agg2<<<(E + 255) / 256, 256, 0, stream>>>(src, dst, dinv, g, agg2, E);
  k_final<<<(N + 255) / 256, 256, 0, stream>>>(agg2, dinv, g, b2, out, N);
}